// CapsuleLayer_42064909697697
// MI455X (gfx1250) — compile-verified
//
#include <hip/hip_runtime.h>
#include <math.h>

typedef __attribute__((ext_vector_type(2))) float v2f;
typedef __attribute__((ext_vector_type(8))) float v8f;

#define T0C   4
#define T1C   8
#define Z0C   16
#define HH    48
#define WWC   48
#define DDC   48
#define KTOT  (Z0C * 27)            // 432
#define CONV_OUT 128                // T1*Z1
#define NSTAGE (KTOT * 16 / 256)    // 27 staging iterations per group
#define FLDS_SIZE (KTOT * 16)       // 6912 floats = 27 KB, WMMA-B fragment layout

__global__ __launch_bounds__(256) void capsule_fused_kernel(
    const float* __restrict__ u,     // [T0, Z0, H, W, D]
    const float* __restrict__ Wt,    // [T0, 128, 432] (k = ((z0*3+kh)*3+kw)*3+kd contiguous)
    const float* __restrict__ bias,  // [T0, 128]
    float* __restrict__ out)         // [128, H, W, D]
{
  // B-fragment layout: element (k, v) at flds[(k>>1)*32 + v*2 + (k&1)].
  // Lane's WMMA-B pair (k0+khalf*2, k0+khalf*2+1) is contiguous 8B:
  //   addr = (khalf*32 + vox*2)*4  +  (k0/2)*128   -> one ds_load_b64, imm offset.
  __shared__ float flds[FLDS_SIZE];
  __shared__ float blds[T1C * 16];

  const int tid   = threadIdx.x;
  const int lane  = tid & 31;
  const int m     = tid >> 5;       // wave id == t1
  const int vox   = lane & 15;      // voxel within d-tile (WMMA N index)
  const int khalf = lane >> 4;      // lane half: selects K pair / z1 upper half

  const int bid = blockIdx.x;
  const int h   = bid / (WWC * 3);
  const int rem = bid % (WWC * 3);
  const int w   = rem / 3;
  const int d0  = (rem % 3) * 16;

  // ---- one-time precompute of per-thread staging slots ----
  // staging iter n writes flds[n*256 + tid]; that slot is element (k, v) with
  //   k = n*16 + (tid>>5)*2 + (tid&1),   v = (tid>>1)&15
  const int sv   = (tid >> 1) & 15;
  const int kvar = ((tid >> 5) << 1) | (tid & 1);
  int goff[NSTAGE];
  unsigned vmask = 0u;
#pragma unroll
  for (int n = 0; n < NSTAGE; ++n) {
    const int k  = n * 16 + kvar;
    const int z0 = k / 27;
    const int r  = k % 27;
    const int kh = r / 9;
    const int r9 = r % 9;
    const int kw = r9 / 3;
    const int kd = r9 % 3;
    const int hh = h + kh - 1;
    const int ww = w + kw - 1;
    const int dd = d0 + sv + kd - 1;
    const bool ok = (unsigned)hh < (unsigned)HH &&
                    (unsigned)ww < (unsigned)WWC &&
                    (unsigned)dd < (unsigned)DDC;
    goff[n] = ok ? (((z0 * HH + hh) * WWC + ww) * DDC + dd) : 0;
    vmask |= (ok ? 1u : 0u) << n;
  }

  v8f accS = {};   // sum over the 4 groups of relu(conv+bias)
  v8f y0v  = {};   // group 0
  v8f y3v  = {};   // group 3

#pragma unroll
  for (int g = 0; g < T0C; ++g) {
    // ---- stage WMMA-B fragment tile for this group ----
    __syncthreads();
    const float* ubase = u + (size_t)g * (Z0C * HH * WWC * DDC);
#pragma unroll
    for (int n = 0; n < NSTAGE; ++n) {
      float val = ubase[goff[n]];                       // clamped-safe address
      val = ((vmask >> n) & 1u) ? val : 0.0f;
      flds[n * 256 + tid] = val;
    }
    __syncthreads();

    // ---- GEMM: 16 channels (this wave) x 16 voxels, K = 432, via f32 WMMA ----
    const float* wrow  = Wt + (size_t)(g * CONV_OUT + m * 16 + vox) * KTOT + khalf * 2;
    const float* fbase = &flds[khalf * 32 + vox * 2];
    v8f acc = {};
#pragma unroll
    for (int k0 = 0; k0 < KTOT; k0 += 4) {
      v2f a_frag = *(const v2f*)(wrow + k0);            // global_load_b64, imm offset
      v2f b_frag = *(const v2f*)(fbase + (k0 >> 1) * 32); // ds_load_b64, imm offset
      acc = __builtin_amdgcn_wmma_f32_16x16x4_f32(
          false, a_frag, false, b_frag, (short)0, acc, false, false);
    }

    // ---- bias + relu, collect group results (g is compile-time) ----
    const float* bptr = bias + g * CONV_OUT + m * 16 + khalf * 8;
#pragma unroll
    for (int r = 0; r < 8; ++r) {
      float y = acc[r] + bptr[r];
      y = y > 0.0f ? y : 0.0f;
      accS[r] += y;
      if (g == 0) y0v[r] = y;
      if (g == 3) y3v[r] = y;
    }
  }

  // Per-lane view: voxel = vox, z1 = khalf*8 + r  (WMMA C layout).
  // z1-reduction = 8-wide register sum + shfl_xor over lane halves (wave32).

  // ---- routing iteration 1: s1 = (1/8) * sum_i y_i ----
  v8f s1, act;
  float sq1 = 0.0f, U3 = 0.0f;
#pragma unroll
  for (int r = 0; r < 8; ++r) {
    s1[r] = accS[r] * 0.125f;
    sq1 += s1[r] * s1[r];
    U3  += y3v[r];
  }
  sq1 += __shfl_xor(sq1, 16, 32);
  U3  += __shfl_xor(U3, 16, 32);
  float scale = sq1 / ((1.0f + sq1) * sqrtf(sq1 + 1e-9f));
  float A1 = 0.0f;
#pragma unroll
  for (int r = 0; r < 8; ++r) { act[r] = s1[r] * scale; A1 += act[r]; }
  A1 += __shfl_xor(A1, 16, 32);
  float b1 = A1 * U3;                 // b after iter 1 (list collapsed to group 3)

  // softmax over t1 (across the 8 waves) via LDS
  __syncthreads();
  if (lane < 16) blds[m * 16 + vox] = b1;
  __syncthreads();
  float mx = -3.4e38f;
  float bv[T1C];
#pragma unroll
  for (int t = 0; t < T1C; ++t) { bv[t] = blds[t * 16 + vox]; mx = fmaxf(mx, bv[t]); }
  float se = 0.0f;
#pragma unroll
  for (int t = 0; t < T1C; ++t) se += expf(bv[t] - mx);
  float c1 = expf(bv[m] - mx) / se;

  // ---- routing iteration 2: s2 = softmax(b)[t1] * y0 ----
  float sq2 = 0.0f, U0 = 0.0f;
#pragma unroll
  for (int r = 0; r < 8; ++r) {
    float s = c1 * y0v[r];
    act[r] = s;
    sq2 += s * s;
    U0  += y0v[r];
  }
  sq2 += __shfl_xor(sq2, 16, 32);
  U0  += __shfl_xor(U0, 16, 32);
  scale = sq2 / ((1.0f + sq2) * sqrtf(sq2 + 1e-9f));
  float A2 = 0.0f;
#pragma unroll
  for (int r = 0; r < 8; ++r) { A2 += act[r] * scale; }
  A2 += __shfl_xor(A2, 16, 32);
  float b2 = b1 + A2 * U0;

  __syncthreads();
  if (lane < 16) blds[m * 16 + vox] = b2;
  __syncthreads();
  mx = -3.4e38f;
#pragma unroll
  for (int t = 0; t < T1C; ++t) { bv[t] = blds[t * 16 + vox]; mx = fmaxf(mx, bv[t]); }
  se = 0.0f;
#pragma unroll
  for (int t = 0; t < T1C; ++t) se += expf(bv[t] - mx);
  float c2 = expf(bv[m] - mx) / se;

  // ---- routing iteration 3: activation = squash(softmax(b2)[t1] * y0) ----
  float sq3 = 0.0f;
#pragma unroll
  for (int r = 0; r < 8; ++r) {
    float s = c2 * y0v[r];
    act[r] = s;
    sq3 += s * s;
  }
  sq3 += __shfl_xor(sq3, 16, 32);
  scale = sq3 / ((1.0f + sq3) * sqrtf(sq3 + 1e-9f));

  // ---- store: out[(t1*16 + z1), h, w, d0+vox] ----
  const size_t sp = (size_t)h * (WWC * DDC) + (size_t)w * DDC + d0 + vox;
#pragma unroll
  for (int r = 0; r < 8; ++r) {
    const int z1 = khalf * 8 + r;
    out[(size_t)(m * 16 + z1) * (HH * WWC * DDC) + sp] = act[r] * scale;
  }
}

extern "C" void kernel_launch(void* const* d_in, const int* in_sizes, int n_in,
                              void* d_out, int out_size, void* d_ws, size_t ws_size,
                              hipStream_t stream) {
  const float* u    = (const float*)d_in[0];
  const float* Wt   = (const float*)d_in[1];
  const float* bias = (const float*)d_in[2];
  float* out = (float*)d_out;
  (void)in_sizes; (void)n_in; (void)out_size; (void)d_ws; (void)ws_size;

  dim3 grid(HH * WWC * 3);   // one WG per (h, w, 16-wide d tile) = 6912
  dim3 block(256);           // 8 waves: one per t1
  capsule_fused_kernel<<<grid, block, 0, stream>>>(u, Wt, bias, out);
}